// KalmanDecomposition_66254165508716
// MI455X (gfx1250) — compile-verified
//
#include <hip/hip_runtime.h>

// CDNA5 (gfx1250) WMMA vector types
typedef __attribute__((ext_vector_type(2))) float v2f;
typedef __attribute__((ext_vector_type(8))) float v8f;

#define KD_L  512   // sequence length (time)
#define KD_C  512   // channels
#define KD_B  64    // batch

// ---------------------------------------------------------------------------
// Kernel 1: build the 512x512 smoother matrix M (trend = M @ y per series).
// M column s = filter+RTS-smoother response to unit impulse e_s, using the
// exact reference recursions (R = Q = 1, P seeded with P0 - Q = 0).
// Grid: 32 blocks x 16 threads; one column per thread; mf scratch in LDS.
// ---------------------------------------------------------------------------
__global__ __launch_bounds__(16) void kd_build_M(float* __restrict__ M) {
    __shared__ float Ksh[KD_L];
    __shared__ float Gsh[KD_L];
    __shared__ float mf[KD_L * 16];   // [t][tid] -> 32 KB

    if (threadIdx.x == 0) {
        float P = 0.0f;                      // P0 - Q  (reference seed)
        for (int t = 0; t < KD_L; ++t) {
            float Pp = P + 1.0f;             // predict (Q = 1)
            float K  = Pp / (Pp + 1.0f);     // gain (R = 1)
            P = (1.0f - K) * Pp;             // filtered covariance Pf[t]
            Ksh[t] = K;
            Gsh[t] = P / (P + 1.0f);         // smoother gain G[t]
        }
    }
    __syncthreads();

    const int s = blockIdx.x * 16 + threadIdx.x;   // impulse position / column

    // Forward filter on y = e_s
    float m = 0.0f;
    for (int t = 0; t < KD_L; ++t) {
        float y = (t == s) ? 1.0f : 0.0f;
        m = m + Ksh[t] * (y - m);
        mf[t * 16 + threadIdx.x] = m;
    }

    // Backward RTS smoother; write column s of M (row-major M[l*512 + s])
    float ms = mf[(KD_L - 1) * 16 + threadIdx.x];
    M[(KD_L - 1) * KD_L + s] = ms;
    for (int t = KD_L - 2; t >= 0; --t) {
        float mt = mf[t * 16 + threadIdx.x];
        ms = mt + Gsh[t] * (ms - mt);
        M[t * KD_L + s] = ms;
    }
}

// ---------------------------------------------------------------------------
// Kernel 2: batched GEMM  trend[b] = M (512x512) @ obs[b] (512x512), plus
// residual = obs - trend.  One wave32 computes a 64(L) x 16(C) output tile
// with V_WMMA_F32_16X16X4_F32 (fp32 in/out, K-loop step 4), reusing each
// B(obs) fragment across 4 A(M) fragments.  M and obs[b] live in L2 (192MB),
// so the kernel is HBM-bound only on obs read + trend/residual writes.
// ---------------------------------------------------------------------------
__global__ __launch_bounds__(256) void kd_gemm(const float* __restrict__ obs,
                                               const float* __restrict__ M,
                                               float* __restrict__ out) {
    const int wave = (blockIdx.x * 256 + threadIdx.x) >> 5;   // 0..16383
    const int lane = threadIdx.x & 31;

    const int b     = wave >> 8;            // 256 waves per batch
    const int rem   = wave & 255;
    const int mBase = (rem >> 5) * 64;      // 8 row-groups of 64
    const int nBase = (rem & 31) * 16;      // 32 col-tiles of 16

    const float* __restrict__ Y = obs + (size_t)b * KD_L * KD_C;
    float* __restrict__ T = out + (size_t)b * KD_L * KD_C;
    float* __restrict__ R = out + (size_t)KD_B * KD_L * KD_C
                                + (size_t)b * KD_L * KD_C;

    const int r    = lane & 15;   // row (A) / col (B,C) within tile
    const int half = lane >> 4;   // lane-half selects K pair / C row offset

    v8f acc0 = {}; v8f acc1 = {}; v8f acc2 = {}; v8f acc3 = {};

    const float* Arow0 = M + (size_t)(mBase      + r) * KD_L;
    const float* Arow1 = M + (size_t)(mBase + 16 + r) * KD_L;
    const float* Arow2 = M + (size_t)(mBase + 32 + r) * KD_L;
    const float* Arow3 = M + (size_t)(mBase + 48 + r) * KD_L;

    for (int k = 0; k < KD_L; k += 4) {
        const int kk = k + 2 * half;        // this lane-half covers K = kk, kk+1

        // B fragment: 4x16 tile of obs[b] at (k, nBase)
        v2f bf;
        bf.x = Y[(size_t)kk * KD_C + nBase + r];
        bf.y = Y[(size_t)(kk + 1) * KD_C + nBase + r];

        // A fragments: 16x4 tiles of M at (mBase + 16*mt, k); contiguous pairs
        v2f a0 = *(const v2f*)(Arow0 + kk);
        v2f a1 = *(const v2f*)(Arow1 + kk);
        v2f a2 = *(const v2f*)(Arow2 + kk);
        v2f a3 = *(const v2f*)(Arow3 + kk);

        acc0 = __builtin_amdgcn_wmma_f32_16x16x4_f32(false, a0, false, bf,
                                                     (short)0, acc0, false, false);
        acc1 = __builtin_amdgcn_wmma_f32_16x16x4_f32(false, a1, false, bf,
                                                     (short)0, acc1, false, false);
        acc2 = __builtin_amdgcn_wmma_f32_16x16x4_f32(false, a2, false, bf,
                                                     (short)0, acc2, false, false);
        acc3 = __builtin_amdgcn_wmma_f32_16x16x4_f32(false, a3, false, bf,
                                                     (short)0, acc3, false, false);
    }

    // Epilogue: C/D layout -> element (row = 8*half + v, col = r) per VGPR v.
    const int c = nBase + r;
#pragma unroll
    for (int v = 0; v < 8; ++v) {
        {
            int l = mBase + 8 * half + v;
            float t = acc0[v]; float y = Y[(size_t)l * KD_C + c];
            T[(size_t)l * KD_C + c] = t;  R[(size_t)l * KD_C + c] = y - t;
        }
        {
            int l = mBase + 16 + 8 * half + v;
            float t = acc1[v]; float y = Y[(size_t)l * KD_C + c];
            T[(size_t)l * KD_C + c] = t;  R[(size_t)l * KD_C + c] = y - t;
        }
        {
            int l = mBase + 32 + 8 * half + v;
            float t = acc2[v]; float y = Y[(size_t)l * KD_C + c];
            T[(size_t)l * KD_C + c] = t;  R[(size_t)l * KD_C + c] = y - t;
        }
        {
            int l = mBase + 48 + 8 * half + v;
            float t = acc3[v]; float y = Y[(size_t)l * KD_C + c];
            T[(size_t)l * KD_C + c] = t;  R[(size_t)l * KD_C + c] = y - t;
        }
    }
}

extern "C" void kernel_launch(void* const* d_in, const int* in_sizes, int n_in,
                              void* d_out, int out_size, void* d_ws, size_t ws_size,
                              hipStream_t stream) {
    (void)in_sizes; (void)n_in; (void)out_size; (void)ws_size;

    const float* obs = (const float*)d_in[0];
    float* out = (float*)d_out;
    float* M   = (float*)d_ws;            // 512*512 floats = 1 MB scratch

    // 1) Build smoother matrix (data-independent, rebuilt every call for
    //    graph-capture determinism).
    kd_build_M<<<KD_L / 16, 16, 0, stream>>>(M);

    // 2) Batched GEMM + residual: 16384 waves, 8 waves (256 threads) / block.
    kd_gemm<<<(KD_B * 8 * 32) / 8, 256, 0, stream>>>(obs, M, out);
}